// cheb_conv_with_SAt_43739946942869
// MI455X (gfx1250) — compile-verified
//
#include <hip/hip_runtime.h>
#include <hip/hip_bf16.h>
#include <math.h>

typedef float v2f __attribute__((ext_vector_type(2)));
typedef float v8f __attribute__((ext_vector_type(8)));

#define T_LEN   4096
#define BATCH   1024
#define NSENS   10
#define JPAD    12               // j dimension padded to 12 (multiple of K=4)
#define XROW    (BATCH * NSENS)  // 10240 floats per t-row of x

// ---------------------------------------------------------------------------
// Kernel 1: 5x5 scaled Laplacian (Jacobi eigensolve for lambda_max),
// Chebyshev polynomials K=3, the 10x10 replicated masks M_k, and an 8-float
// zero pad area used by padding lanes in stage B. Single thread.
// ---------------------------------------------------------------------------
__global__ void prep_cheb_kernel(const float* __restrict__ adj,
                                 float* __restrict__ M /* [3][10][10] */,
                                 float* __restrict__ zpad /* [8] */) {
  if (threadIdx.x != 0 || blockIdx.x != 0) return;

  for (int z = 0; z < 8; ++z) zpad[z] = 0.f;

  float L[5][5], A[5][5];
  for (int r = 0; r < 5; ++r) {
    float rs = 0.f;
    for (int c = 0; c < 5; ++c) rs += adj[r * 5 + c];
    for (int c = 0; c < 5; ++c)
      L[r][c] = (r == c ? rs : 0.f) - adj[r * 5 + c];
  }
  for (int r = 0; r < 5; ++r)
    for (int c = 0; c < 5; ++c) A[r][c] = L[r][c];

  // Cyclic Jacobi: diagonalize symmetric 5x5, take max eigenvalue.
  for (int sweep = 0; sweep < 30; ++sweep) {
    for (int p = 0; p < 4; ++p) {
      for (int q = p + 1; q < 5; ++q) {
        float apq = A[p][q];
        if (fabsf(apq) < 1e-12f) continue;
        float theta = (A[q][q] - A[p][p]) / (2.f * apq);
        float t = 1.f / (fabsf(theta) + sqrtf(theta * theta + 1.f));
        if (theta < 0.f) t = -t;
        float c = 1.f / sqrtf(t * t + 1.f);
        float s = t * c;
        for (int k = 0; k < 5; ++k) {
          float apk = A[p][k], aqk = A[q][k];
          A[p][k] = c * apk - s * aqk;
          A[q][k] = s * apk + c * aqk;
        }
        for (int k = 0; k < 5; ++k) {
          float akp = A[k][p], akq = A[k][q];
          A[k][p] = c * akp - s * akq;
          A[k][q] = s * akp + c * akq;
        }
      }
    }
  }
  float lam = A[0][0];
  for (int r = 1; r < 5; ++r) lam = fmaxf(lam, A[r][r]);

  float Lt[5][5];
  for (int r = 0; r < 5; ++r)
    for (int c = 0; c < 5; ++c)
      Lt[r][c] = 2.f * L[r][c] / lam - (r == c ? 1.f : 0.f);

  float C[3][5][5];
  for (int r = 0; r < 5; ++r)
    for (int c = 0; c < 5; ++c) {
      C[0][r][c] = (r == c ? 1.f : 0.f);
      C[1][r][c] = Lt[r][c];
    }
  for (int r = 0; r < 5; ++r)
    for (int c = 0; c < 5; ++c) {
      float acc = 0.f;
      for (int m = 0; m < 5; ++m) acc += Lt[r][m] * C[1][m][c];
      C[2][r][c] = 2.f * acc - C[0][r][c];
    }

  // M_k[r][c] = C[k][((r*10+c)%25)/5][(r*10+c)%5]
  for (int k = 0; k < 3; ++k)
    for (int r = 0; r < 10; ++r)
      for (int c = 0; c < 10; ++c) {
        int p = (r * 10 + c) % 25;
        M[k * 100 + r * 10 + c] = C[k][p / 5][p % 5];
      }
}

// ---------------------------------------------------------------------------
// Kernel 2: W[i][j][f] = sum_k M_k[j][i] * Theta[k][j][f], j padded to 12.
// ---------------------------------------------------------------------------
__global__ void prep_w_kernel(const float* __restrict__ M,
                              const float* __restrict__ Theta, /* [3][10][1024] */
                              float* __restrict__ Wt /* [10][12][1024] */) {
  int idx = blockIdx.x * blockDim.x + threadIdx.x;
  if (idx >= NSENS * JPAD * BATCH) return;
  int f = idx & (BATCH - 1);
  int j = (idx >> 10) % JPAD;
  int i = idx / (JPAD * BATCH);
  float v = 0.f;
  if (j < NSENS) {
    for (int k = 0; k < 3; ++k)
      v += M[k * 100 + j * 10 + i] * Theta[k * (NSENS * BATCH) + j * BATCH + f];
  }
  Wt[idx] = v;
}

// ---------------------------------------------------------------------------
// Stage B: S_i[t, j] = sum_b x[t, b, i] * sa[b, j, i]  via V_WMMA_F32_16X16X4_F32
// Block = 20 waves: wave w -> sensor i = w%10, K-half = w/10 (b split 2x512 for
// wave-level parallelism). Padding lanes (j >= 10) read an 8-float zero area
// with stride 0 -> branchless inner loop, no exec-mask churn.
// Half tiles are reduced through LDS (identical D-fragment mapping).
// ---------------------------------------------------------------------------
__global__ void __launch_bounds__(640)
stage_b_kernel(const float* __restrict__ x,    /* [4096][1024][10] */
               const float* __restrict__ sa,   /* [1024][10][10]   */
               const float* __restrict__ zpad, /* [8] zeros        */
               float* __restrict__ S /* [10][4096][12] */) {
  __shared__ float red[NSENS * 256]; // one 16x16 f32 tile per sensor

  const int tile_t = blockIdx.x;        // 0..255
  const int wave   = threadIdx.x >> 5;  // 0..19
  const int i      = wave % NSENS;      // sensor index
  const int half   = wave / NSENS;      // K-split half (0 or 1)
  const int lane   = threadIdx.x & 31;
  const int t0     = tile_t * 16;
  const int m      = lane & 15;         // A row
  const int hi     = lane >> 4;         // half-wave selector (K packing)
  const int n      = lane & 15;         // B column (= j)

  const int b_base = half * (BATCH / 2) + (hi << 1); // starting b for this lane

  // A operand pointer: x[t0+m, b, i]; advances 4 b's (40 floats) per step.
  const float* xp = x + (size_t)(t0 + m) * XROW + (size_t)b_base * NSENS + i;

  // B operand pointers: sa[b, n, i] and sa[b+1, n, i]; padding lanes use the
  // zero area with stride 0 (stays in L0, costs nothing, no branches).
  const bool valid = (n < NSENS);
  const float* p0 = valid ? (sa + (size_t)b_base * 100 + n * 10 + i) : zpad;
  const float* p1 = valid ? (p0 + 100) : zpad;
  const int    bstep = valid ? 400 : 0; // 4 b's * 100 floats

  v8f acc = {0.f, 0.f, 0.f, 0.f, 0.f, 0.f, 0.f, 0.f};

#pragma unroll 4
  for (int it = 0; it < BATCH / 2 / 4; ++it) { // 128 K-steps of K=4
    v2f a, b;
    a.x = xp[0];
    a.y = xp[NSENS];
    b.x = p0[0];
    b.y = p1[0];
    acc = __builtin_amdgcn_wmma_f32_16x16x4_f32(false, a, false, b,
                                                (short)0, acc, false, false);
    xp += 4 * NSENS;
    p0 += bstep;
    p1 += bstep;
  }

  // D 16x16 f32 layout: VGPR r, lanes 0-15 -> (M=r, N=n); lanes 16-31 -> (M=r+8, N=n).
  // Reduce the two K-halves through LDS; half 1 writes the final S tile.
  const int frag = (hi << 3) * 16 + n; // (row = r + hi*8) * 16 + col
  if (half == 0) {
#pragma unroll
    for (int r = 0; r < 8; ++r) red[i * 256 + frag + r * 16] = acc[r];
  }
  __syncthreads();
  if (half == 1 && n < JPAD) {
    float* Sp = S + ((size_t)i * T_LEN + t0) * JPAD + n;
#pragma unroll
    for (int r = 0; r < 8; ++r) {
      int trow = r + (hi << 3);
      Sp[(size_t)trow * JPAD] = acc[r] + red[i * 256 + frag + r * 16];
    }
  }
}

// ---------------------------------------------------------------------------
// Stage C: out[t, f, i] = relu( sum_j S_i[t, j] * W_i[j, f] ), 3 WMMAs per tile.
// Block = 10 waves (one per i) on one (t,f) 16x16 tile; results staged in LDS
// and written as contiguous 160-float rows -> fully coalesced 168 MB of stores.
// ---------------------------------------------------------------------------
__global__ void __launch_bounds__(320)
stage_c_kernel(const float* __restrict__ S,  /* [10][4096][12] */
               const float* __restrict__ Wt, /* [10][12][1024] */
               float* __restrict__ out /* [4096][1024][10] */) {
  __shared__ float tile[16 * 160];

  const int tile_f = blockIdx.x & 63;   // 64 f-tiles
  const int tile_t = blockIdx.x >> 6;   // 256 t-tiles
  const int i      = threadIdx.x >> 5;
  const int lane   = threadIdx.x & 31;
  const int t0 = tile_t * 16;
  const int f0 = tile_f * 16;
  const int m  = lane & 15;
  const int hi = lane >> 4;
  const int n  = lane & 15;

  v8f acc = {0.f, 0.f, 0.f, 0.f, 0.f, 0.f, 0.f, 0.f};

#pragma unroll
  for (int j0 = 0; j0 < JPAD; j0 += 4) {
    const int kj = j0 + (hi << 1);
    v2f a;
    const float* sp = S + ((size_t)i * T_LEN + (t0 + m)) * JPAD + kj;
    a.x = sp[0];
    a.y = sp[1];
    v2f b;
    const float* wp = Wt + ((size_t)i * JPAD + kj) * BATCH + f0 + n;
    b.x = wp[0];
    b.y = wp[BATCH];
    acc = __builtin_amdgcn_wmma_f32_16x16x4_f32(false, a, false, b,
                                                (short)0, acc, false, false);
  }

  // relu + transpose (t, f, i) into LDS
#pragma unroll
  for (int r = 0; r < 8; ++r) {
    int trow = r + (hi << 3);
    tile[trow * 160 + n * NSENS + i] = fmaxf(acc[r], 0.f);
  }
  __syncthreads();

  // Coalesced store: 16 rows x 160 contiguous floats; 320 threads x 8 elems.
  for (int idx = threadIdx.x; idx < 16 * 160; idx += 320) {
    int tr = idx / 160;
    int c  = idx - tr * 160;
    out[(size_t)(t0 + tr) * XROW + (size_t)f0 * NSENS + c] = tile[idx];
  }
}

// ---------------------------------------------------------------------------
// Host-side launcher
// ---------------------------------------------------------------------------
extern "C" void kernel_launch(void* const* d_in, const int* in_sizes, int n_in,
                              void* d_out, int out_size, void* d_ws, size_t ws_size,
                              hipStream_t stream) {
  const float* x     = (const float*)d_in[0]; // (4096, 1024, 10)
  const float* adj   = (const float*)d_in[1]; // (5, 5)
  const float* sa    = (const float*)d_in[2]; // (1024, 10, 10)
  const float* Theta = (const float*)d_in[3]; // (3, 10, 1024)
  float* out = (float*)d_out;                 // (4096, 1024, 10, 1)

  char* ws = (char*)d_ws;
  float* M    = (float*)(ws);           // 3*100 floats        (1200 B)
  float* zpad = (float*)(ws + 2048);    // 8 floats of zeros
  float* Wt   = (float*)(ws + 4096);    // 10*12*1024 floats   (491,520 B)
  float* S    = (float*)(ws + 524288);  // 10*4096*12 floats   (1,966,080 B)

  prep_cheb_kernel<<<1, 64, 0, stream>>>(adj, M, zpad);

  int wtot = NSENS * JPAD * BATCH; // 122,880
  prep_w_kernel<<<(wtot + 255) / 256, 256, 0, stream>>>(M, Theta, Wt);

  stage_b_kernel<<<T_LEN / 16, 640, 0, stream>>>(x, sa, zpad, S);

  stage_c_kernel<<<(T_LEN / 16) * (BATCH / 16), 320, 0, stream>>>(S, Wt, out);
}